// CAGAM_88098369176224
// MI455X (gfx1250) — compile-verified
//
#include <hip/hip_runtime.h>
#include <hip/hip_bf16.h>

typedef __attribute__((ext_vector_type(2))) float v2f;
typedef __attribute__((ext_vector_type(8))) float v8f;

#define S_STRIDE 1028   // 16-row S strip stride (floats); rows offset by 4 banks
#define RSQRT6 0.40824829046386301637f

// ---------------------------------------------------------------------------
// Kernel 1: context = relu(x * Wc + bc); Q/K/V = 6x6 convs of context.
// x: [32,512,1024], ctx: [32,6,1024], Qf/Kf: [32*1024][8] (pad 6,7 = 0),
// Vf: [32*1024][6] compact (V never feeds WMMA, no K-pad needed).
// Memory-bound: each x element read once, 128B coalesced per wave per channel.
// ---------------------------------------------------------------------------
__global__ __launch_bounds__(256) void ctx_qkv_kernel(
    const float* __restrict__ x,
    const float* __restrict__ cw, const float* __restrict__ cb,
    const float* __restrict__ qw, const float* __restrict__ qb,
    const float* __restrict__ kw, const float* __restrict__ kb,
    const float* __restrict__ vw, const float* __restrict__ vb,
    float* __restrict__ ctx, float* __restrict__ Qf,
    float* __restrict__ Kf, float* __restrict__ Vf)
{
    __shared__ float wlds[6 * 512];
    for (int i = threadIdx.x; i < 6 * 512; i += 256) wlds[i] = cw[i];
    __syncthreads();

    const int g = blockIdx.x * 256 + threadIdx.x;     // global position 0..32767
    const int b = g >> 10;
    const int p = g & 1023;
    const float* xp = x + ((size_t)b * 512) * 1024 + p;

    float acc[6] = {0.f, 0.f, 0.f, 0.f, 0.f, 0.f};
#pragma unroll 4
    for (int c = 0; c < 512; ++c) {
        const float xv = xp[(size_t)c * 1024];
#pragma unroll
        for (int i = 0; i < 6; ++i) acc[i] += xv * wlds[i * 512 + c];
    }

    float ct[6];
#pragma unroll
    for (int i = 0; i < 6; ++i) ct[i] = fmaxf(acc[i] + cb[i], 0.f);

    float q[6], k[6], v[6];
#pragma unroll
    for (int o = 0; o < 6; ++o) {
        float aq = qb[o], ak = kb[o], av = vb[o];
#pragma unroll
        for (int i = 0; i < 6; ++i) {
            aq += qw[o * 6 + i] * ct[i];
            ak += kw[o * 6 + i] * ct[i];
            av += vw[o * 6 + i] * ct[i];
        }
        q[o] = aq; k[o] = ak; v[o] = av;
    }

#pragma unroll
    for (int i = 0; i < 6; ++i) ctx[((size_t)b * 6 + i) * 1024 + p] = ct[i];

    float4* Qr = (float4*)(Qf + (size_t)g * 8);
    float4* Kr = (float4*)(Kf + (size_t)g * 8);
    Qr[0] = make_float4(q[0], q[1], q[2], q[3]); Qr[1] = make_float4(q[4], q[5], 0.f, 0.f);
    Kr[0] = make_float4(k[0], k[1], k[2], k[3]); Kr[1] = make_float4(k[4], k[5], 0.f, 0.f);

    float2* Vr = (float2*)(Vf + (size_t)g * 6);
    Vr[0] = make_float2(v[0], v[1]);
    Vr[1] = make_float2(v[2], v[3]);
    Vr[2] = make_float2(v[4], v[5]);
}

// ---------------------------------------------------------------------------
// Kernel 3: cam bilinear 8x8 -> 32x32 (half-pixel) + 6x6 convs -> Qd/Kd rows.
// ---------------------------------------------------------------------------
__global__ __launch_bounds__(256) void cam_kernel(
    const float* __restrict__ cam,
    const float* __restrict__ cqw, const float* __restrict__ cqb,
    const float* __restrict__ ckw, const float* __restrict__ ckb,
    float* __restrict__ Qdf, float* __restrict__ Kdf)
{
    const int g = blockIdx.x * 256 + threadIdx.x;
    const int b = g >> 10;
    const int p = g & 1023;
    const int oy = p >> 5, ox = p & 31;

    const float sy = (oy + 0.5f) * 0.25f - 0.5f;
    const float sx = (ox + 0.5f) * 0.25f - 0.5f;
    const int y0 = (int)floorf(sy); const float fy = sy - (float)y0;
    const int x0 = (int)floorf(sx); const float fx = sx - (float)x0;
    const int y0c = min(max(y0, 0), 7), y1c = min(max(y0 + 1, 0), 7);
    const int x0c = min(max(x0, 0), 7), x1c = min(max(x0 + 1, 0), 7);

    const float* cbase = cam + (size_t)b * 6 * 64;
    float cv[6];
#pragma unroll
    for (int i = 0; i < 6; ++i) {
        const float* ci = cbase + i * 64;
        const float v00 = ci[y0c * 8 + x0c], v01 = ci[y0c * 8 + x1c];
        const float v10 = ci[y1c * 8 + x0c], v11 = ci[y1c * 8 + x1c];
        cv[i] = (1.f - fy) * ((1.f - fx) * v00 + fx * v01)
              +         fy * ((1.f - fx) * v10 + fx * v11);
    }

    float qd[6], kd[6];
#pragma unroll
    for (int o = 0; o < 6; ++o) {
        float aq = cqb[o], ak = ckb[o];
#pragma unroll
        for (int i = 0; i < 6; ++i) {
            aq += cqw[o * 6 + i] * cv[i];
            ak += ckw[o * 6 + i] * cv[i];
        }
        qd[o] = aq; kd[o] = ak;
    }

    float4* Qr = (float4*)(Qdf + (size_t)g * 8);
    float4* Kr = (float4*)(Kdf + (size_t)g * 8);
    Qr[0] = make_float4(qd[0], qd[1], qd[2], qd[3]); Qr[1] = make_float4(qd[4], qd[5], 0.f, 0.f);
    Kr[0] = make_float4(kd[0], kd[1], kd[2], kd[3]); Kr[1] = make_float4(kd[4], kd[5], 0.f, 0.f);
}

// ---------------------------------------------------------------------------
// Kernel 2: fused dual-GEMM + softmax + P*V. One WG = one batch x 128 rows.
// Phase 1 (per 16-row sub-block): 8 waves compute 16x16 S tiles with
// v_wmma_f32_16x16x4_f32 (K=6 padded to 8 -> 2 k-steps), S = Pd*Pnd/sqrt(6),
// spilled to LDS in C-layout. Phase 2: 16 lanes/row do softmax + 6-dim V
// accumulation with half-wave shfl_xor reductions. No [1024,1024] HBM traffic.
// LDS total 155,904 B < 160 KB -> two workgroups resident per 320 KB WGP.
// ---------------------------------------------------------------------------
__global__ __launch_bounds__(256) void attn_kernel(
    const float* __restrict__ Qf, const float* __restrict__ Kf,
    const float* __restrict__ Vf, const float* __restrict__ Qdf,
    const float* __restrict__ Kdf, float* __restrict__ enh)
{
    extern __shared__ float smem[];
    float* sK  = smem;            // 1024*8 = 8192 floats
    float* sKd = smem + 8192;     // 1024*8
    float* sV  = smem + 16384;    // 1024*6 = 6144 floats (compact)
    float* sS  = smem + 22528;    // 16 * S_STRIDE

    const int tid = threadIdx.x;
    const int b = blockIdx.x >> 3;
    const int n0 = (blockIdx.x & 7) * 128;
    const size_t rowBase = (size_t)b * 1024;

    { // stage K, Kd ([pos][8]) and V ([pos][6]) for this batch into LDS
        const float4* gK  = (const float4*)(Kf  + rowBase * 8);
        const float4* gKd = (const float4*)(Kdf + rowBase * 8);
        const float4* gV  = (const float4*)(Vf  + rowBase * 6);
        float4* lK = (float4*)sK; float4* lKd = (float4*)sKd; float4* lV = (float4*)sV;
        for (int i = tid; i < 2048; i += 256) { lK[i] = gK[i]; lKd[i] = gKd[i]; }
        for (int i = tid; i < 1536; i += 256) { lV[i] = gV[i]; }
    }
    __syncthreads();

    const int lane  = tid & 31;
    const int wave  = tid >> 5;
    const int idx16 = lane & 15;         // M (A) / N (B,C) index
    const int hi    = lane >> 4;         // lane half
    // f32 WMMA layouts: vgpr0 = {K0 | K2}, vgpr1 = {K1 | K3} per k-step
    const int sel0 = hi ? 2 : 0, sel1 = hi ? 3 : 1;
    const int sel2 = hi ? 6 : 4, sel3 = hi ? 7 : 5;   // K=6,7 are zero-padded

    const int r2  = tid >> 4;            // phase-2 row 0..15
    const int c16 = tid & 15;            // phase-2 lane-in-row

    for (int s = 0; s < 8; ++s) {
        const int n0s = n0 + s * 16;
        // ---------------- phase 1: WMMA S tiles ----------------
        {
            const float* Qr  = Qf  + (rowBase + n0s + idx16) * 8;
            const float* Qdr = Qdf + (rowBase + n0s + idx16) * 8;
            v2f aq0 = { Qr[sel0],  Qr[sel1]  };
            v2f aq1 = { Qr[sel2],  Qr[sel3]  };
            v2f ad0 = { Qdr[sel0], Qdr[sel1] };
            v2f ad1 = { Qdr[sel2], Qdr[sel3] };

            for (int t = wave; t < 64; t += 8) {
                const int m0 = t * 16;
                const float* Kr  = sK  + (m0 + idx16) * 8;
                const float* Kdr = sKd + (m0 + idx16) * 8;
                v2f bk0 = { Kr[sel0],  Kr[sel1]  };
                v2f bk1 = { Kr[sel2],  Kr[sel3]  };
                v2f bd0 = { Kdr[sel0], Kdr[sel1] };
                v2f bd1 = { Kdr[sel2], Kdr[sel3] };

                v8f cz = {0.f, 0.f, 0.f, 0.f, 0.f, 0.f, 0.f, 0.f};
                v8f cnd = __builtin_amdgcn_wmma_f32_16x16x4_f32(
                    false, aq0, false, bk0, (short)0, cz, false, false);
                cnd = __builtin_amdgcn_wmma_f32_16x16x4_f32(
                    false, aq1, false, bk1, (short)0, cnd, false, false);
                v8f cd = __builtin_amdgcn_wmma_f32_16x16x4_f32(
                    false, ad0, false, bd0, (short)0, cz, false, false);
                cd = __builtin_amdgcn_wmma_f32_16x16x4_f32(
                    false, ad1, false, bd1, (short)0, cd, false, false);

                // C layout: vgpr r -> row r (lanes 0-15) / r+8 (lanes 16-31), col = lane%16
#pragma unroll
                for (int r = 0; r < 8; ++r) {
                    const float sv = cnd[r] * cd[r] * RSQRT6;
                    sS[(r + (hi ? 8 : 0)) * S_STRIDE + m0 + idx16] = sv;
                }
            }
        }
        __syncthreads();
        // ---------------- phase 2: softmax + P*V ----------------
        {
            const float* Srow = sS + r2 * S_STRIDE;
            float mx = -3.402823466e38f;
            for (int m = c16; m < 1024; m += 16) mx = fmaxf(mx, Srow[m]);
#pragma unroll
            for (int off = 8; off > 0; off >>= 1) mx = fmaxf(mx, __shfl_xor(mx, off, 16));

            float sum = 0.f;
            float a0 = 0.f, a1 = 0.f, a2 = 0.f, a3 = 0.f, a4 = 0.f, a5 = 0.f;
            for (int m = c16; m < 1024; m += 16) {
                const float e = __expf(Srow[m] - mx);
                sum += e;
                const float2* Vr = (const float2*)(sV + m * 6);
                const float2 v0 = Vr[0]; const float2 v1 = Vr[1]; const float2 v2 = Vr[2];
                a0 += e * v0.x; a1 += e * v0.y; a2 += e * v1.x;
                a3 += e * v1.y; a4 += e * v2.x; a5 += e * v2.y;
            }
#pragma unroll
            for (int off = 8; off > 0; off >>= 1) {
                sum += __shfl_xor(sum, off, 16);
                a0 += __shfl_xor(a0, off, 16); a1 += __shfl_xor(a1, off, 16);
                a2 += __shfl_xor(a2, off, 16); a3 += __shfl_xor(a3, off, 16);
                a4 += __shfl_xor(a4, off, 16); a5 += __shfl_xor(a5, off, 16);
            }
            if (c16 == 0) {
                const float inv = 1.0f / sum;
                const int n = n0s + r2;
                float* e0 = enh + (size_t)b * 6 * 1024 + n;
                e0[0]        = a0 * inv; e0[1024]     = a1 * inv;
                e0[2 * 1024] = a2 * inv; e0[3 * 1024] = a3 * inv;
                e0[4 * 1024] = a4 * inv; e0[5 * 1024] = a5 * inv;
            }
        }
        __syncthreads();
    }
}

// ---------------------------------------------------------------------------
// Kernel 4: enhanced = ctx + beta*enh; output_maps = 15x6 conv.
// ---------------------------------------------------------------------------
__global__ __launch_bounds__(256) void output_kernel(
    const float* __restrict__ ctx, const float* __restrict__ enh,
    const float* __restrict__ ow, const float* __restrict__ ob,
    const float* __restrict__ beta, float* __restrict__ out)
{
    const int g = blockIdx.x * 256 + threadIdx.x;
    const int b = g >> 10;
    const int p = g & 1023;
    const float bt = beta[0];

    float e[6];
#pragma unroll
    for (int i = 0; i < 6; ++i) {
        const size_t idx = ((size_t)b * 6 + i) * 1024 + p;
        e[i] = ctx[idx] + bt * enh[idx];
    }
#pragma unroll
    for (int o = 0; o < 15; ++o) {
        float a = ob[o];
#pragma unroll
        for (int i = 0; i < 6; ++i) a += ow[o * 6 + i] * e[i];
        out[((size_t)b * 15 + o) * 1024 + p] = a;
    }
}

// ---------------------------------------------------------------------------
// Kernel 5: logits[b,o] = mean over HW. One block per (b,o), no atomics.
// ---------------------------------------------------------------------------
__global__ __launch_bounds__(256) void logits_kernel(
    const float* __restrict__ maps, float* __restrict__ logits)
{
    __shared__ float red[256];
    const float* r = maps + (size_t)blockIdx.x * 1024;
    float s = 0.f;
    for (int i = threadIdx.x; i < 1024; i += 256) s += r[i];
    red[threadIdx.x] = s;
    __syncthreads();
    for (int o = 128; o > 0; o >>= 1) {
        if (threadIdx.x < o) red[threadIdx.x] += red[threadIdx.x + o];
        __syncthreads();
    }
    if (threadIdx.x == 0) logits[blockIdx.x] = red[0] * (1.0f / 1024.0f);
}

// ---------------------------------------------------------------------------
extern "C" void kernel_launch(void* const* d_in, const int* in_sizes, int n_in,
                              void* d_out, int out_size, void* d_ws, size_t ws_size,
                              hipStream_t stream) {
    const float* x    = (const float*)d_in[0];
    const float* cam  = (const float*)d_in[1];
    const float* cw   = (const float*)d_in[2];
    const float* cb   = (const float*)d_in[3];
    const float* qw   = (const float*)d_in[4];
    const float* qb   = (const float*)d_in[5];
    const float* kw   = (const float*)d_in[6];
    const float* kb   = (const float*)d_in[7];
    const float* vw   = (const float*)d_in[8];
    const float* vb   = (const float*)d_in[9];
    const float* cqw  = (const float*)d_in[10];
    const float* cqb  = (const float*)d_in[11];
    const float* ckw  = (const float*)d_in[12];
    const float* ckb  = (const float*)d_in[13];
    const float* ow   = (const float*)d_in[14];
    const float* ob   = (const float*)d_in[15];
    const float* beta = (const float*)d_in[16];

    float* ws  = (float*)d_ws;
    float* Qf  = ws;                  // 32*1024*8 = 262144
    float* Kf  = ws + 262144;         // 32*1024*8
    float* Qdf = ws + 524288;         // 32*1024*8
    float* Kdf = ws + 786432;         // 32*1024*8
    float* Vf  = ws + 1048576;        // 32*1024*6 = 196608 (compact)
    float* ctx = ws + 1245184;        // 32*6*1024
    float* enh = ws + 1441792;        // 32*6*1024

    float* out_maps   = (float*)d_out;            // 32*15*1024
    float* out_logits = (float*)d_out + 491520;   // 32*15

    ctx_qkv_kernel<<<128, 256, 0, stream>>>(x, cw, cb, qw, qb, kw, kb, vw, vb,
                                            ctx, Qf, Kf, Vf);
    cam_kernel<<<128, 256, 0, stream>>>(cam, cqw, cqb, ckw, ckb, Qdf, Kdf);

    const size_t smem = (size_t)(22528 + 16 * S_STRIDE) * sizeof(float); // 155,904 B
    attn_kernel<<<256, 256, smem, stream>>>(Qf, Kf, Vf, Qdf, Kdf, enh);

    output_kernel<<<128, 256, 0, stream>>>(ctx, enh, ow, ob, beta, out_maps);
    logits_kernel<<<480, 256, 0, stream>>>(out_maps, out_logits);
}